// UrbanGNN_78340203479715
// MI455X (gfx1250) — compile-verified
//
#include <hip/hip_runtime.h>
#include <hip/hip_bf16.h>
#include <math.h>

typedef __attribute__((ext_vector_type(2))) float v2f;
typedef __attribute__((ext_vector_type(8))) float v8f;

// ---------------------------------------------------------------------------
// fp32 WMMA GEMM, compile-time K/M:  D[n,M] = A[n,K] @ W[K,M]
//   - one wave computes a 16 x (16*NT) output strip
//   - K multiple of 4 (layer-1 inputs are zero-padded to K=12)
//   - A fragment loaded once per K-step (b64), reused for NT column tiles
// ---------------------------------------------------------------------------
template <int K, int M, int NT>
__global__ void __launch_bounds__(256)
gemm_t(const float* __restrict__ A, const float* __restrict__ W,
       float* __restrict__ D, int n) {
  static_assert(K % 4 == 0 && M % (16 * NT) == 0, "shape");
  const int lane = threadIdx.x & 31;
  const int wave = threadIdx.x >> 5;
  constexpr int GR = M / (16 * NT);            // column groups
  const int tilesN = n >> 4;
  const int tile = blockIdx.x * 8 + wave;
  if (tile >= tilesN * GR) return;             // wave-uniform exit
  const int tr = tile / GR;
  const int tg = tile % GR;
  const int hl = lane & 15;
  const int kb = (lane >> 4) << 1;             // 0 (lanes 0-15) / 2 (lanes 16-31)
  const int row = tr * 16 + hl;
  const float* __restrict__ Arow = A + (size_t)row * K + kb;

  int cols[NT];
  v8f acc[NT] = {};
#pragma unroll
  for (int j = 0; j < NT; ++j) cols[j] = tg * (16 * NT) + j * 16 + hl;

#pragma unroll 8
  for (int k0 = 0; k0 < K; k0 += 4) {
    const v2f a = *(const v2f*)(Arow + k0);    // rows K=k0+kb, k0+kb+1
#pragma unroll
    for (int j = 0; j < NT; ++j) {
      v2f b;
      b.x = W[(size_t)(k0 + kb) * M + cols[j]];
      b.y = W[(size_t)(k0 + kb + 1) * M + cols[j]];
      acc[j] = __builtin_amdgcn_wmma_f32_16x16x4_f32(
          false, a, false, b, (short)0, acc[j], false, false);
    }
  }
  const int r0 = tr * 16 + ((lane >> 4) << 3);
#pragma unroll
  for (int j = 0; j < NT; ++j)
#pragma unroll
    for (int i = 0; i < 8; ++i)
      D[(size_t)(r0 + i) * M + cols[j]] = acc[j][i];
}

// ---------------------------------------------------------------------------
// Helpers
// ---------------------------------------------------------------------------
__device__ __forceinline__ unsigned fenc(float f) {   // order-preserving key
  unsigned u = __float_as_uint(f);
  return (u & 0x80000000u) ? ~u : (u | 0x80000000u);
}
__device__ __forceinline__ float fdec(unsigned u) {
  return __uint_as_float((u & 0x80000000u) ? (u ^ 0x80000000u) : ~u);
}
__device__ __forceinline__ void edge_sd(const long long* __restrict__ ei,
                                        long long E, long long e, int& s, int& d) {
  if (e < E) { s = (int)ei[e]; d = (int)ei[E + e]; }
  else       { s = d = (int)(e - E); }            // self loops appended
}

__global__ void zero_f32(float* p, long long cnt) {
  long long i = (long long)blockIdx.x * blockDim.x + threadIdx.x;
  if (i < cnt) p[i] = 0.f;
}

// zero-pad x[n,10] -> xp[n,12]
__global__ void pad_x(const float* __restrict__ x, float* __restrict__ xp, int n) {
  int i = blockIdx.x * blockDim.x + threadIdx.x;
  if (i >= n * 12) return;
  int r = i / 12, k = i % 12;
  xp[i] = (k < 10) ? x[r * 10 + k] : 0.f;
}
// zero-pad W1[10,256] -> W1p[12,256]
__global__ void pad_w(const float* __restrict__ w, float* __restrict__ wp) {
  int i = blockIdx.x * blockDim.x + threadIdx.x;
  if (i >= 12 * 256) return;
  int k = i / 256;
  wp[i] = (k < 10) ? w[i] : 0.f;
}

// e_src[n,H], e_dst[n,H] from h[n,H*C]
__global__ void att_coef(const float* __restrict__ h,
                         const float* __restrict__ a_src,
                         const float* __restrict__ a_dst,
                         float* __restrict__ esrc, float* __restrict__ edst,
                         int n, int H, int C) {
  int i = blockIdx.x * blockDim.x + threadIdx.x;
  if (i >= n * H) return;
  int node = i / H, hh = i % H;
  const float* hp = h + (size_t)node * H * C + (size_t)hh * C;
  const float* as = a_src + hh * C;
  const float* ad = a_dst + hh * C;
  float ss = 0.f, sd = 0.f;
  for (int c = 0; c < C; ++c) { float v = hp[c]; ss += v * as[c]; sd += v * ad[c]; }
  esrc[i] = ss; edst[i] = sd;
}

// pass 1: e = leakyrelu(e_src[src]+e_dst[dst]); segment max over dst
__global__ void edge_logits_max(const long long* __restrict__ ei, long long E,
                                long long Etot,
                                const float* __restrict__ esrc,
                                const float* __restrict__ edst,
                                float* __restrict__ ebuf,
                                unsigned* __restrict__ mEnc, int H) {
  long long idx = (long long)blockIdx.x * blockDim.x + threadIdx.x;
  if (idx >= Etot * H) return;
  long long e = idx / H; int hh = (int)(idx % H);
  int s, d; edge_sd(ei, E, e, s, d);
  float v = esrc[s * H + hh] + edst[d * H + hh];
  v = (v >= 0.f) ? v : 0.2f * v;
  ebuf[idx] = v;
  atomicMax(&mEnc[d * H + hh], fenc(v));
}

// pass 2: ex = exp(e - m[dst]); segment sum over dst
__global__ void edge_exp_sum(const long long* __restrict__ ei, long long E,
                             long long Etot,
                             const unsigned* __restrict__ mEnc,
                             float* __restrict__ ebuf,
                             float* __restrict__ ssum, int H) {
  long long idx = (long long)blockIdx.x * blockDim.x + threadIdx.x;
  if (idx >= Etot * H) return;
  long long e = idx / H; int hh = (int)(idx % H);
  int s, d; edge_sd(ei, E, e, s, d);
  float m = fdec(mEnc[d * H + hh]);
  float ex = __expf(ebuf[idx] - m);
  ebuf[idx] = ex;
  atomicAdd(&ssum[d * H + hh], ex);
}

// pass 3: out[dst] += h[src] * alpha   (one thread per edge x 4 channels)
__global__ void gat_scatter4(const long long* __restrict__ ei, long long E,
                             long long Etot,
                             const float* __restrict__ h,
                             const float* __restrict__ ebuf,
                             const float* __restrict__ ssum,
                             float* __restrict__ out, int H, int C) {
  const int HC = H * C;
  const int HCq = HC >> 2;
  long long idx = (long long)blockIdx.x * blockDim.x + threadIdx.x;
  if (idx >= Etot * (long long)HCq) return;
  long long e = idx / HCq;
  int r = (int)(idx % HCq) << 2;               // channel group start
  int hh = r / C;                              // head (group never spans heads)
  int s, d; edge_sd(ei, E, e, s, d);
  float alpha = ebuf[e * H + hh] / (ssum[d * H + hh] + 1e-16f);
  const float4 hv = *(const float4*)(h + (size_t)s * HC + r);
  float* op = out + (size_t)d * HC + r;
  atomicAdd(op + 0, hv.x * alpha);
  atomicAdd(op + 1, hv.y * alpha);
  atomicAdd(op + 2, hv.z * alpha);
  atomicAdd(op + 3, hv.w * alpha);
}

__global__ void bias_elu(float* __restrict__ x, const float* __restrict__ b,
                         long long total, int M) {
  long long i = (long long)blockIdx.x * blockDim.x + threadIdx.x;
  if (i >= total) return;
  float v = x[i] + b[(int)(i % M)];
  x[i] = (v > 0.f) ? v : expm1f(v);
}

__global__ void deg_kernel(const long long* __restrict__ ei, long long E,
                           long long Etot, float* __restrict__ deg) {
  long long idx = (long long)blockIdx.x * blockDim.x + threadIdx.x;
  if (idx >= Etot) return;
  int d = (idx < E) ? (int)ei[E + idx] : (int)(idx - E);
  atomicAdd(&deg[d], 1.f);
}

__global__ void dinv_kernel(const float* __restrict__ deg,
                            float* __restrict__ dinv, int n) {
  int i = blockIdx.x * blockDim.x + threadIdx.x;
  if (i < n) dinv[i] = rsqrtf(fmaxf(deg[i], 1e-12f));
}

__global__ void gcn_scatter4(const long long* __restrict__ ei, long long E,
                             long long Etot,
                             const float* __restrict__ h2,
                             const float* __restrict__ dinv,
                             float* __restrict__ out, int M) {
  const int Mq = M >> 2;
  long long idx = (long long)blockIdx.x * blockDim.x + threadIdx.x;
  if (idx >= Etot * (long long)Mq) return;
  long long e = idx / Mq;
  int c = (int)(idx % Mq) << 2;
  int s, d; edge_sd(ei, E, e, s, d);
  float nrm = dinv[s] * dinv[d];
  const float4 hv = *(const float4*)(h2 + (size_t)s * M + c);
  float* op = out + (size_t)d * M + c;
  atomicAdd(op + 0, hv.x * nrm);
  atomicAdd(op + 1, hv.y * nrm);
  atomicAdd(op + 2, hv.z * nrm);
  atomicAdd(op + 3, hv.w * nrm);
}

// column-sum of h[n,32] into gsum[32]
__global__ void __launch_bounds__(256)
mean_reduce(const float* __restrict__ h, float* __restrict__ gsum, int n) {
  __shared__ float sh[256];
  const int col  = threadIdx.x & 31;
  const int rgrp = threadIdx.x >> 5;   // 0..7
  float acc = 0.f;
  for (int r = blockIdx.x * 8 + rgrp; r < n; r += gridDim.x * 8)
    acc += h[(size_t)r * 32 + col];
  sh[threadIdx.x] = acc;
  __syncthreads();
  if (threadIdx.x < 32) {
    float s = 0.f;
    for (int i = threadIdx.x; i < 256; i += 32) s += sh[i];
    atomicAdd(&gsum[threadIdx.x], s);
  }
}

__global__ void zone_kernel(const float* __restrict__ h,
                            const float* __restrict__ Wz,
                            const float* __restrict__ bz,
                            float* __restrict__ out, int n) {
  int i = blockIdx.x * blockDim.x + threadIdx.x;
  if (i >= n) return;
  float acc0 = bz[0], acc1 = bz[1], acc2 = bz[2], acc3 = bz[3];
  for (int c = 0; c < 32; ++c) {
    float v = h[(size_t)i * 32 + c];
    acc0 += v * Wz[c * 4 + 0];
    acc1 += v * Wz[c * 4 + 1];
    acc2 += v * Wz[c * 4 + 2];
    acc3 += v * Wz[c * 4 + 3];
  }
  out[(size_t)i * 4 + 0] = acc0;
  out[(size_t)i * 4 + 1] = acc1;
  out[(size_t)i * 4 + 2] = acc2;
  out[(size_t)i * 4 + 3] = acc3;
}

__global__ void heads_kernel(const float* __restrict__ gsum, float inv_n,
                             const float* __restrict__ Ws, const float* __restrict__ bs,
                             const float* __restrict__ Wa, const float* __restrict__ ba,
                             const float* __restrict__ Wc, const float* __restrict__ bc,
                             float* __restrict__ out3) {
  float s = bs[0], a = ba[0], c = bc[0];
  for (int i = 0; i < 32; ++i) {
    float ge = gsum[i] * inv_n;
    s += ge * Ws[i];
    a += ge * Wa[i];
    c += ge * Wc[i];
  }
  out3[0] = 1.f / (1.f + __expf(-s));
  out3[1] = 1.f / (1.f + __expf(-a));
  out3[2] = 1.f / (1.f + __expf(-c));
}

// ---------------------------------------------------------------------------
static inline int nblk(long long t) { return (int)((t + 255) / 256); }

template <int K, int M, int NT>
static void launch_gemm(const float* A, const float* W, float* D, int n,
                        hipStream_t stream) {
  long long tiles = (long long)(n >> 4) * (M / (16 * NT));
  gemm_t<K, M, NT><<<(int)((tiles + 7) / 8), 256, 0, stream>>>(A, W, D, n);
}

extern "C" void kernel_launch(void* const* d_in, const int* in_sizes, int n_in,
                              void* d_out, int out_size, void* d_ws, size_t ws_size,
                              hipStream_t stream) {
  const float*     x   = (const float*)d_in[0];
  const long long* ei  = (const long long*)d_in[1];
  const float* W1  = (const float*)d_in[2];
  const float* a1s = (const float*)d_in[3];
  const float* a1d = (const float*)d_in[4];
  const float* b1  = (const float*)d_in[5];
  const float* W2  = (const float*)d_in[6];
  const float* a2s = (const float*)d_in[7];
  const float* a2d = (const float*)d_in[8];
  const float* b2  = (const float*)d_in[9];
  const float* W3  = (const float*)d_in[10];
  const float* b3  = (const float*)d_in[11];
  const float* W4  = (const float*)d_in[12];
  const float* b4  = (const float*)d_in[13];
  const float* Wz  = (const float*)d_in[14];
  const float* bz  = (const float*)d_in[15];
  const float* Ws  = (const float*)d_in[16];
  const float* bs  = (const float*)d_in[17];
  const float* Wa  = (const float*)d_in[18];
  const float* ba  = (const float*)d_in[19];
  const float* Wc  = (const float*)d_in[20];
  const float* bc  = (const float*)d_in[21];

  const int n = in_sizes[0] / 10;            // 50000 (multiple of 16)
  const long long E    = in_sizes[1] / 2;    // 400000
  const long long Etot = E + n;              // with self loops

  // workspace layout (floats)
  float* ws = (float*)d_ws;
  size_t off = 0;
  float* bufA = ws + off;               off += (size_t)n * 256;   // gemm outputs
  float* bufB = ws + off;               off += (size_t)n * 256;   // aggregation
  float* xp   = ws + off;               off += (size_t)n * 12;    // padded x
  float* W1p  = ws + off;               off += (size_t)12 * 256;  // padded W1
  float* esrc = ws + off;               off += (size_t)n * 4;
  float* edst = ws + off;               off += (size_t)n * 4;
  unsigned* mEnc = (unsigned*)(ws + off); off += (size_t)n * 4;
  float* ssum = ws + off;               off += (size_t)n * 4;
  float* ebuf = ws + off;               off += (size_t)Etot * 4;
  float* deg  = ws + off;               off += (size_t)n;
  float* dinv = ws + off;               off += (size_t)n;
  float* gsum = ws + off;               off += 32;

  float* outv   = (float*)d_out;
  float* zoneo  = outv;                              // [n,4]
  float* heads3 = outv + (size_t)n * 4;              // 3 scalars
  float* hfin   = outv + (size_t)n * 4 + 3;          // [n,32]

  // ---------------- GAT layer 1: 10 -> 4 heads x 64 -------------------------
  pad_x<<<nblk((long long)n * 12), 256, 0, stream>>>(x, xp, n);
  pad_w<<<nblk(12 * 256), 256, 0, stream>>>(W1, W1p);
  launch_gemm<12, 256, 2>(xp, W1p, bufA, n, stream);
  att_coef<<<nblk((long long)n * 4), 256, 0, stream>>>(bufA, a1s, a1d, esrc, edst, n, 4, 64);
  zero_f32<<<nblk((long long)n * 4), 256, 0, stream>>>((float*)mEnc, (long long)n * 4);
  zero_f32<<<nblk((long long)n * 4), 256, 0, stream>>>(ssum, (long long)n * 4);
  zero_f32<<<nblk((long long)n * 256), 256, 0, stream>>>(bufB, (long long)n * 256);
  edge_logits_max<<<nblk(Etot * 4), 256, 0, stream>>>(ei, E, Etot, esrc, edst, ebuf, mEnc, 4);
  edge_exp_sum<<<nblk(Etot * 4), 256, 0, stream>>>(ei, E, Etot, mEnc, ebuf, ssum, 4);
  gat_scatter4<<<nblk(Etot * 64), 256, 0, stream>>>(ei, E, Etot, bufA, ebuf, ssum, bufB, 4, 64);
  bias_elu<<<nblk((long long)n * 256), 256, 0, stream>>>(bufB, b1, (long long)n * 256, 256);

  // ---------------- GAT layer 2: 256 -> 2 heads x 64 ------------------------
  launch_gemm<256, 128, 2>(bufB, W2, bufA, n, stream);
  float* out2 = bufA + (size_t)n * 128;              // second half of bufA
  att_coef<<<nblk((long long)n * 2), 256, 0, stream>>>(bufA, a2s, a2d, esrc, edst, n, 2, 64);
  zero_f32<<<nblk((long long)n * 2), 256, 0, stream>>>((float*)mEnc, (long long)n * 2);
  zero_f32<<<nblk((long long)n * 2), 256, 0, stream>>>(ssum, (long long)n * 2);
  zero_f32<<<nblk((long long)n * 128), 256, 0, stream>>>(out2, (long long)n * 128);
  edge_logits_max<<<nblk(Etot * 2), 256, 0, stream>>>(ei, E, Etot, esrc, edst, ebuf, mEnc, 2);
  edge_exp_sum<<<nblk(Etot * 2), 256, 0, stream>>>(ei, E, Etot, mEnc, ebuf, ssum, 2);
  gat_scatter4<<<nblk(Etot * 32), 256, 0, stream>>>(ei, E, Etot, bufA, ebuf, ssum, out2, 2, 64);
  bias_elu<<<nblk((long long)n * 128), 256, 0, stream>>>(out2, b2, (long long)n * 128, 128);

  // ---------------- degree normalization (shared by both GCN layers) --------
  zero_f32<<<nblk((long long)n), 256, 0, stream>>>(deg, (long long)n);
  deg_kernel<<<nblk(Etot), 256, 0, stream>>>(ei, E, Etot, deg);
  dinv_kernel<<<nblk((long long)n), 256, 0, stream>>>(deg, dinv, n);

  // ---------------- GCN layer 1: 128 -> 64 ----------------------------------
  launch_gemm<128, 64, 2>(out2, W3, bufB, n, stream);
  float* out3 = bufB + (size_t)n * 64;
  zero_f32<<<nblk((long long)n * 64), 256, 0, stream>>>(out3, (long long)n * 64);
  gcn_scatter4<<<nblk(Etot * 16), 256, 0, stream>>>(ei, E, Etot, bufB, dinv, out3, 64);
  bias_elu<<<nblk((long long)n * 64), 256, 0, stream>>>(out3, b3, (long long)n * 64, 64);

  // ---------------- GCN layer 2: 64 -> 32 (writes final h into d_out) -------
  launch_gemm<64, 32, 2>(out3, W4, bufA, n, stream);
  zero_f32<<<nblk((long long)n * 32), 256, 0, stream>>>(hfin, (long long)n * 32);
  gcn_scatter4<<<nblk(Etot * 8), 256, 0, stream>>>(ei, E, Etot, bufA, dinv, hfin, 32);
  bias_elu<<<nblk((long long)n * 32), 256, 0, stream>>>(hfin, b4, (long long)n * 32, 32);

  // ---------------- heads ----------------------------------------------------
  zero_f32<<<1, 32, 0, stream>>>(gsum, 32);
  mean_reduce<<<128, 256, 0, stream>>>(hfin, gsum, n);
  zone_kernel<<<nblk((long long)n), 256, 0, stream>>>(hfin, Wz, bz, zoneo, n);
  heads_kernel<<<1, 1, 0, stream>>>(gsum, 1.f / (float)n, Ws, bs, Wa, ba, Wc, bc, heads3);
}